// AVSL_Similarity_52278341927639
// MI455X (gfx1250) — compile-verified
//
#include <hip/hip_runtime.h>
#include <hip/hip_bf16.h>
#include <math.h>

typedef __attribute__((ext_vector_type(2))) float v2f;
typedef __attribute__((ext_vector_type(8))) float v8f;

#define DDIM 512
#define B1N  256
#define B2N  512
#define LN   3
#define STR  513   // padded LDS row stride (odd -> low bank conflicts)

// ---------------- Kernel 1: row-wise L2 normalize (x / max(||x||, 1e-12)) ----
__global__ __launch_bounds__(256) void norm_rows_kernel(const float* __restrict__ src,
                                                        float* __restrict__ dst,
                                                        int nrows) {
    int row = blockIdx.x;
    if (row >= nrows) return;
    const float* x = src + (size_t)row * DDIM;
    float* y = dst + (size_t)row * DDIM;
    int t = threadIdx.x;
    float a = x[t];
    float b = x[t + 256];
    float ss = a * a + b * b;
    // wave32 reduce
    for (int off = 16; off > 0; off >>= 1) ss += __shfl_down(ss, off, 32);
    __shared__ float wsum[8];
    int lane = t & 31, wv = t >> 5;
    if (lane == 0) wsum[wv] = ss;
    __syncthreads();
    float tot = 0.f;
#pragma unroll
    for (int i = 0; i < 8; ++i) tot += wsum[i];
    float s = 1.0f / fmaxf(sqrtf(tot), 1e-12f);
    y[t] = a * s;
    y[t + 256] = b * s;
}

// ---------------- Kernel 2: top-3 per column of each link matrix ------------
// Produces normalized weights w_j = v_j / (v0+v1+v2 + 1e-8) and row indices.
__global__ __launch_bounds__(256) void topk_cols_kernel(const float* __restrict__ links,
                                                        float* __restrict__ wout,
                                                        int* __restrict__ iout) {
    int c = blockIdx.x * blockDim.x + threadIdx.x;   // 0 .. 2*512-1
    if (c >= (LN - 1) * DDIM) return;
    int lm = c / DDIM;
    int col = c % DDIM;
    const float* W = links + (size_t)lm * DDIM * DDIM;
    float v0 = -INFINITY, v1 = -INFINITY, v2 = -INFINITY;
    int i0 = 0, i1 = 0, i2 = 0;
    for (int r = 0; r < DDIM; ++r) {
        float v = W[(size_t)r * DDIM + col];
        if (v > v0)      { v2 = v1; i2 = i1; v1 = v0; i1 = i0; v0 = v; i0 = r; }
        else if (v > v1) { v2 = v1; i2 = i1; v1 = v;  i1 = r; }
        else if (v > v2) { v2 = v;  i2 = r; }
    }
    float inv = 1.0f / (v0 + v1 + v2 + 1e-8f);
    int base = c * 3;
    wout[base]     = v0 * inv;
    wout[base + 1] = v1 * inv;
    wout[base + 2] = v2 * inv;
    iout[base]     = i0;
    iout[base + 1] = i1;
    iout[base + 2] = i2;
}

// ---------------- Kernel 3: main chained-similarity pipeline ----------------
// One block = (b1, 16 consecutive b2). 16 threads per pair, 32 k's per thread.
// nh tile lives in LDS; layers 1..2 gather 3 sparse taps + sigmoid blend.
// Final sum over k: all 8 waves run V_WMMA_F32_16X16X4_F32 (B = ones) over
// their own 64-k slice in parallel, then 16 threads combine the partials.
__global__ __launch_bounds__(256) void avsl_main_kernel(
    const float* __restrict__ e1n, const float* __restrict__ e2n,
    const float* __restrict__ cert1, const float* __restrict__ cert2,
    const float* __restrict__ tkw, const int* __restrict__ tki,
    const float* __restrict__ alpha, const float* __restrict__ beta,
    float* __restrict__ out) {
    __shared__ float nh[16 * STR];
    __shared__ float part[8][16];

    int t = threadIdx.x;
    int b1 = blockIdx.y;
    int b2base = blockIdx.x * 16;
    int p = t >> 4;          // pair within tile: 0..15
    int l16 = t & 15;        // k-lane within pair
    int b2 = b2base + p;

    // prefetch next layer's operands into cache (global_prefetch_b8 path)
    __builtin_prefetch(e1n + (size_t)(B1N + b1) * DDIM, 0, 1);
    __builtin_prefetch(e2n + (size_t)(B2N + b2) * DDIM, 0, 1);
    __builtin_prefetch(cert1 + (size_t)(B1N + b1) * DDIM, 0, 1);
    __builtin_prefetch(cert2 + (size_t)(B2N + b2) * DDIM, 0, 1);

    // ----- layer 0: nodes_hat = (e1-e2)^2 -----
    {
        const float* e1 = e1n + (size_t)b1 * DDIM;
        const float* e2 = e2n + (size_t)b2 * DDIM;
#pragma unroll 8
        for (int i = 0; i < 32; ++i) {
            int k = l16 + 16 * i;
            float dv = e1[k] - e2[k];
            nh[p * STR + k] = dv * dv;
        }
    }
    __syncthreads();

    // ----- layers 1..2: sparse propagate + sigmoid rectify -----
    float nv[32];
    for (int l = 1; l < LN; ++l) {
        const float* e1l = e1n + (size_t)(l * B1N + b1) * DDIM;
        const float* e2l = e2n + (size_t)(l * B2N + b2) * DDIM;
        const float* c1l = cert1 + (size_t)(l * B1N + b1) * DDIM;
        const float* c2l = cert2 + (size_t)(l * B2N + b2) * DDIM;
        const float* al = alpha + (size_t)(l - 1) * DDIM;
        const float* be = beta + (size_t)(l - 1) * DDIM;
        const float* w3 = tkw + (size_t)(l - 1) * DDIM * 3;
        const int* i3 = tki + (size_t)(l - 1) * DDIM * 3;
#pragma unroll 4
        for (int i = 0; i < 32; ++i) {
            int k = l16 + 16 * i;
            int kb = k * 3;
            int ia = i3[kb], ib = i3[kb + 1], ic = i3[kb + 2];
            float acc = nh[p * STR + ia] * w3[kb]
                      + nh[p * STR + ib] * w3[kb + 1]
                      + nh[p * STR + ic] * w3[kb + 2];
            float eta = c1l[k] * c2l[k];
            float xin = al[k] * eta + be[k];
            float Pv = 1.0f / (1.0f + __expf(-xin));
            float dv = e1l[k] - e2l[k];
            float nodes = dv * dv;
            nv[i] = (1.0f - Pv) * acc + Pv * nodes;
        }
        __syncthreads();
#pragma unroll 8
        for (int i = 0; i < 32; ++i) nh[p * STR + l16 + 16 * i] = nv[i];
        __syncthreads();
    }

    // ----- final reduction over k via f32 WMMA, all 8 waves in parallel -----
    {
        int wv = t >> 5;                  // wave id: 0..7, owns k in [wv*64, wv*64+64)
        int lane = t & 31;
        int row = lane & 15;              // A-matrix M index (pair)
        int koff = (lane >> 4) * 2;       // lanes 0-15 -> K+0/+1, 16-31 -> K+2/+3
        v2f bm; bm.x = 1.0f; bm.y = 1.0f; // B = ones(4x16)
        v8f c;
#pragma unroll
        for (int j = 0; j < 8; ++j) c[j] = 0.0f;
        int kbase = wv * 64;
#pragma unroll 4
        for (int kc = kbase; kc < kbase + 64; kc += 4) {
            v2f a;
            a.x = nh[row * STR + kc + koff];
            a.y = nh[row * STR + kc + koff + 1];
            c = __builtin_amdgcn_wmma_f32_16x16x4_f32(
                /*neg_a=*/false, a, /*neg_b=*/false, bm,
                /*c_mod=*/(short)0, c, /*reuse_a=*/false, /*reuse_b=*/false);
        }
        // D[m,n] = partial sum for pair m (all n equal). Column n=0 lives in
        // lane 0 (M=0..7) and lane 16 (M=8..15), one M per accumulator VGPR.
        if (lane == 0) {
#pragma unroll
            for (int j = 0; j < 8; ++j) part[wv][j] = c[j];
        } else if (lane == 16) {
#pragma unroll
            for (int j = 0; j < 8; ++j) part[wv][8 + j] = c[j];
        }
    }
    __syncthreads();
    if (t < 16) {
        float s = 0.f;
#pragma unroll
        for (int w = 0; w < 8; ++w) s += part[w][t];
        out[(size_t)b1 * B2N + b2base + t] = s;
    }
}

extern "C" void kernel_launch(void* const* d_in, const int* in_sizes, int n_in,
                              void* d_out, int out_size, void* d_ws, size_t ws_size,
                              hipStream_t stream) {
    (void)in_sizes; (void)n_in; (void)out_size; (void)ws_size;
    const float* emb1  = (const float*)d_in[0];
    const float* cert1 = (const float*)d_in[1];
    const float* emb2  = (const float*)d_in[2];
    const float* cert2 = (const float*)d_in[3];
    const float* links = (const float*)d_in[4];
    const float* alpha = (const float*)d_in[5];
    const float* beta  = (const float*)d_in[6];

    float* ws  = (float*)d_ws;
    float* e1n = ws;                                   // 3*256*512
    float* e2n = e1n + (size_t)LN * B1N * DDIM;        // 3*512*512
    float* tkw = e2n + (size_t)LN * B2N * DDIM;        // 2*512*3
    int*   tki = (int*)(tkw + (size_t)(LN - 1) * DDIM * 3);

    norm_rows_kernel<<<LN * B1N, 256, 0, stream>>>(emb1, e1n, LN * B1N);
    norm_rows_kernel<<<LN * B2N, 256, 0, stream>>>(emb2, e2n, LN * B2N);
    topk_cols_kernel<<<((LN - 1) * DDIM + 255) / 256, 256, 0, stream>>>(links, tkw, tki);

    dim3 grid(B2N / 16, B1N);
    avsl_main_kernel<<<grid, 256, 0, stream>>>(e1n, e2n, cert1, cert2,
                                               tkw, tki, alpha, beta, (float*)d_out);
}